// DLUPack_72181220376850
// MI455X (gfx1250) — compile-verified
//
#include <hip/hip_runtime.h>
#include <hip/hip_bf16.h>

typedef __attribute__((ext_vector_type(2))) float v2f;
typedef __attribute__((ext_vector_type(8))) float v8f;

#define N_   4
#define C_   256
#define H_   64
#define W_   64
#define S_   2
#define K2_  25
#define CK_  64
#define HW_  (H_ * W_)      // 4096
#define HO_  (H_ * S_)      // 128
#define WO_  (W_ * S_)      // 128
#define NOUT 33             // 8 offset channels + 25 mask channels

// ---------------------------------------------------------------------------
// Kernel A: 1x1 channel compressor as an fp32 WMMA GEMM.
// Per image: cx[ck, p] = sum_c W[ck, c] * x[c, p] + b[ck]
// M=64, N=4096, K=256. One wave computes a 16x16 (ck x pixel) tile via
// V_WMMA_F32_16X16X4_F32, K-loop of 64 steps of 4.
//
// A 16x4 f32 layout (ISA 7.12.2): lanes 0-15 carry M=lane, VGPR0=K0/VGPR1=K1;
// lanes 16-31 carry M=lane-16, VGPR0=K2/VGPR1=K3.
// B 4x16 f32: rows striped across lane halves with stride nrows/2 (mirrors
// the C/D convention: VGPR0 = rows {0,2}, VGPR1 = rows {1,3}).
// C/D 16x16 f32: VGPR r -> row r (lanes 0-15) / row r+8 (lanes 16-31).
// ---------------------------------------------------------------------------
__global__ void compress_wmma(const float* __restrict__ x,
                              const float* __restrict__ wc,
                              const float* __restrict__ bc,
                              float* __restrict__ cx) {
    int wave = blockIdx.x * 4 + (threadIdx.x >> 5);
    int lane = threadIdx.x & 31;
    int lm   = lane & 15;     // M (A) / column (B, C/D)
    int lh   = lane >> 4;     // lane-half selector

    int n  = wave >> 10;               // 1024 waves per image
    int t  = wave & 1023;
    int m0 = (t >> 8) << 4;            // ck tile base  (0..48)
    int p0 = (t & 255) << 4;           // pixel tile base (0..4080)

    const float* xb = x + n * (C_ * HW_);

    v8f acc = {};
    for (int k0 = 0; k0 < C_; k0 += 4) {
        int ka = k0 + 2 * lh;
        v2f a, b;
        a.x = wc[(m0 + lm) * C_ + ka];         // A: W[m, ka]
        a.y = wc[(m0 + lm) * C_ + ka + 1];     // A: W[m, ka+1]
        b.x = xb[(ka    ) * HW_ + p0 + lm];    // B: x[ka,   col]
        b.y = xb[(ka + 1) * HW_ + p0 + lm];    // B: x[ka+1, col]
        acc = __builtin_amdgcn_wmma_f32_16x16x4_f32(
            /*neg_a=*/false, a, /*neg_b=*/false, b,
            /*c_mod=*/(short)0, acc, /*reuse_a=*/false, /*reuse_b=*/false);
    }

#pragma unroll
    for (int r = 0; r < 8; ++r) {
        int ck = m0 + r + 8 * lh;
        cx[(n * CK_ + ck) * HW_ + p0 + lm] = acc[r] + bc[ck];
    }
}

// ---------------------------------------------------------------------------
// Kernel B: fused 3x3 convs (offset: 8 ch, mask: 25 ch) + softmax over the
// 25 mask taps. All 33*64*9 = 19008 weights (76 KB) staged in LDS (gfx1250
// allows up to 320 KB per workgroup). One thread per pixel; inner loop reads
// one cx value and FMAs into 33 accumulators with broadcast LDS weight reads
// (conflict-free).
// ---------------------------------------------------------------------------
__global__ void convs_softmax(const float* __restrict__ cx,
                              const float* __restrict__ wk, const float* __restrict__ bk,
                              const float* __restrict__ wo, const float* __restrict__ bo,
                              float* __restrict__ off_raw,
                              float* __restrict__ mask_sm) {
    extern __shared__ float wlds[];                  // [33][64][9]
    const int NW = NOUT * CK_ * 9;                   // 19008
    for (int e = threadIdx.x; e < NW; e += blockDim.x)
        wlds[e] = (e < 8 * CK_ * 9) ? wo[e] : wk[e - 8 * CK_ * 9];
    __syncthreads();

    int pid = blockIdx.x * blockDim.x + threadIdx.x; // 0..16383
    int n  = pid >> 12;
    int hw = pid & 4095;
    int h  = hw >> 6, w = hw & 63;

    float acc[NOUT];
#pragma unroll
    for (int o = 0; o < 8; ++o)    acc[o] = bo[o];
#pragma unroll
    for (int o = 8; o < NOUT; ++o) acc[o] = bk[o - 8];

    for (int c = 0; c < CK_; ++c) {
        const float* cxc = cx + (n * CK_ + c) * HW_;
#pragma unroll
        for (int tt = 0; tt < 9; ++tt) {
            int dy = tt / 3 - 1, dx = tt % 3 - 1;
            int hh = h + dy, ww = w + dx;
            float v = 0.0f;
            if (hh >= 0 && hh < H_ && ww >= 0 && ww < W_)
                v = cxc[hh * W_ + ww];
            const float* wl = wlds + c * 9 + tt;
#pragma unroll
            for (int o = 0; o < NOUT; ++o)
                acc[o] = fmaf(v, wl[o * (CK_ * 9)], acc[o]);
        }
    }

    // softmax over the 25 mask channels
    float mx = acc[8];
#pragma unroll
    for (int o = 9; o < NOUT; ++o) mx = fmaxf(mx, acc[o]);
    float ex[K2_], sum = 0.0f;
#pragma unroll
    for (int k = 0; k < K2_; ++k) { ex[k] = __expf(acc[8 + k] - mx); sum += ex[k]; }
    float inv = 1.0f / sum;

#pragma unroll
    for (int o = 0; o < 8; ++o)
        off_raw[(n * 8 + o) * HW_ + hw] = acc[o];
#pragma unroll
    for (int k = 0; k < K2_; ++k)
        mask_sm[(n * K2_ + k) * HW_ + hw] = ex[k] * inv;
}

// ---------------------------------------------------------------------------
// Kernel C: fused offset decode + bilinear mask sampling + 25-tap CARAFE.
// The normalized-grid algebra collapses to pixel-space:
//   u  = (y&1)*2 + (x&1)
//   gx = clamp((x>>1) + off_raw[n, u,   y>>1, x>>1], 0, 63)
//   gy = clamp((y>>1) + off_raw[n, 4+u, y>>1, x>>1], 0, 63)
// Block = one output row (n,y), thread = x column. Tap indices/weights are
// precomputed per thread (OOB -> weight 0, matching zero padding), then the
// 256-channel loop does 25 L2-served gathers + FMAs and a coalesced store.
// ---------------------------------------------------------------------------
__global__ void sample_carafe(const float* __restrict__ x,
                              const float* __restrict__ off_raw,
                              const float* __restrict__ mask_sm,
                              float* __restrict__ out) {
    int b  = blockIdx.x;        // 0..511  (n*128 + y)
    int n  = b >> 7;
    int y  = b & 127;
    int xc = threadIdx.x;       // 0..127
    int h0 = y >> 1, w0 = xc >> 1;
    int u  = ((y & 1) << 1) | (xc & 1);
    int hw0 = h0 * W_ + w0;

    float ox = off_raw[(n * 8 + u    ) * HW_ + hw0];
    float oy = off_raw[(n * 8 + 4 + u) * HW_ + hw0];
    float gx = fminf(fmaxf((float)w0 + ox, 0.0f), 63.0f);
    float gy = fminf(fmaxf((float)h0 + oy, 0.0f), 63.0f);
    int x0 = (int)floorf(gx), y0 = (int)floorf(gy);
    int x1 = min(x0 + 1, 63),  y1 = min(y0 + 1, 63);
    float fx = gx - (float)x0, fy = gy - (float)y0;
    float w00 = (1.0f - fx) * (1.0f - fy), w01 = fx * (1.0f - fy);
    float w10 = (1.0f - fx) * fy,          w11 = fx * fy;

    float wgt[K2_];
    int   base[K2_];
#pragma unroll
    for (int k = 0; k < K2_; ++k) {
        const float* mk = mask_sm + (n * K2_ + k) * HW_;
        float mv = w00 * mk[y0 * W_ + x0] + w01 * mk[y0 * W_ + x1]
                 + w10 * mk[y1 * W_ + x0] + w11 * mk[y1 * W_ + x1];
        int i = k / 5, j = k % 5;
        int hy = h0 + i - 2, wx = w0 + j - 2;
        bool ok = (hy >= 0) && (hy < H_) && (wx >= 0) && (wx < W_);
        wgt[k]  = ok ? mv : 0.0f;
        base[k] = ok ? (hy * W_ + wx) : 0;
    }

    const float* xn = x + n * (C_ * HW_);
    float* on = out + ((size_t)(n * C_) * HO_ + y) * WO_ + xc;
#pragma unroll 4
    for (int c = 0; c < C_; ++c) {
        const float* xp = xn + c * HW_;
        float acc = 0.0f;
#pragma unroll
        for (int k = 0; k < K2_; ++k)
            acc = fmaf(xp[base[k]], wgt[k], acc);
        on[(size_t)c * (HO_ * WO_)] = acc;
    }
}

// ---------------------------------------------------------------------------
extern "C" void kernel_launch(void* const* d_in, const int* in_sizes, int n_in,
                              void* d_out, int out_size, void* d_ws, size_t ws_size,
                              hipStream_t stream) {
    const float* x  = (const float*)d_in[0];   // [4,256,64,64]
    const float* wc = (const float*)d_in[1];   // [64,256,1,1]
    const float* bc = (const float*)d_in[2];   // [64]
    const float* wk = (const float*)d_in[3];   // [25,64,3,3]
    const float* bk = (const float*)d_in[4];   // [25]
    const float* wo = (const float*)d_in[5];   // [8,64,3,3]
    const float* bo = (const float*)d_in[6];   // [8]
    float* out = (float*)d_out;                // [4,256,128,128]

    char* ws = (char*)d_ws;
    float* cx      = (float*)ws;                                   // 4 MB
    float* off_raw = (float*)(ws + (size_t)N_ * CK_ * HW_ * 4);    // 512 KB
    float* mask_sm = (float*)(ws + (size_t)N_ * CK_ * HW_ * 4
                                 + (size_t)N_ * 8   * HW_ * 4);    // 1.6 MB

    // A: 4096 waves (4 imgs x 4 ck-tiles x 256 pixel-tiles), 4 waves/block
    compress_wmma<<<1024, 128, 0, stream>>>(x, wc, bc, cx);
    // B: one thread per low-res pixel, 76 KB dynamic LDS for weights
    convs_softmax<<<64, 256, NOUT * CK_ * 9 * sizeof(float), stream>>>(
        cx, wk, bk, wo, bo, off_raw, mask_sm);
    // C: one block per output row (n,y), one thread per x column
    sample_carafe<<<N_ * HO_, WO_, 0, stream>>>(x, off_raw, mask_sm, out);
}